// MITREHeteroGNN_6588479832491
// MI455X (gfx1250) — compile-verified
//
#include <hip/hip_runtime.h>

// ---------------------------------------------------------------------------
// MITRE HeteroGNN forward for gfx1250 (MI455X).
//
// Input leaf order (jax pytree = sorted dict keys):
//   [0..37]   edges: 19 sorted edge keys x (src,dst) int32
//   [38..46]  feats: 9 sorted node types, float32 [N,d]
//   [47..198] convs: layer0 then layer1; per layer 19 sorted edge keys x
//             (W[128,128], att_dst[4,32], att_src[4,32], bias[128])
//   [199..216] enc: 9 sorted node types x (W[d,128], b[128])
//   [217..220] head: l1.W[128,64], l1.b[64], l2.W[64,10], l2.b[10]
// Output: float32 [60000,10].
// ---------------------------------------------------------------------------

typedef float v2f __attribute__((ext_vector_type(2)));
typedef float v8f __attribute__((ext_vector_type(8)));

__device__ __forceinline__ unsigned enc_ord(float f) {
  unsigned u = __float_as_uint(f);
  return (u & 0x80000000u) ? ~u : (u | 0x80000000u);
}
__device__ __forceinline__ float dec_ord(unsigned u) {
  return (u & 0x80000000u) ? __uint_as_float(u & 0x7FFFFFFFu)
                           : __uint_as_float(~u);
}
__device__ __forceinline__ float lrelu(float v) { return v >= 0.f ? v : 0.2f * v; }

// ---------------------------------------------------------------------------
// WMMA f32 GEMM: D[M,N] = A[M,K] @ B[K,N] (+bias) (+relu). Row-major.
// One wave = one 16x16 tile via V_WMMA_F32_16X16X4_F32, K fully unrolled.
// Tail handling: out-of-range rows/cols CLAMP their load addresses (a garbage
// A-row only affects the D-row of the same index, which we never store; same
// for B columns) -> branchless inner loop, guards only on the final stores.
// Two accumulators break the D->C serial chain between consecutive WMMAs.
// ---------------------------------------------------------------------------
template <int K>
__global__ __launch_bounds__(32) void gemm_wmma_f32_kernel(
    const float* __restrict__ A, const float* __restrict__ B,
    const float* __restrict__ bias, float* __restrict__ D,
    int M, int N, int do_relu) {
  const int lane = threadIdx.x;        // 0..31
  const int half = lane >> 4;          // selects K pair / M half
  const int l15  = lane & 15;
  const int row0 = blockIdx.x * 16;
  const int col  = blockIdx.y * 16 + l15;   // real output column
  const int rowA = (row0 + l15 < M) ? (row0 + l15) : (M - 1);  // clamped
  const int colB = (col < N) ? col : (N - 1);                  // clamped

  // A: lane holds A[rowA][k0 + 2*half + {0,1}]  (8B-aligned: strides even)
  const float* Ap  = A + (size_t)rowA * K + 2 * half;
  // B: lane holds B[k0 + 2*half + {0,1}][colB]
  const float* Bp0 = B + (size_t)(2 * half) * N + colB;
  const float* Bp1 = Bp0 + N;
  const size_t bstep = (size_t)8 * N;

  v8f acc0 = {0.f, 0.f, 0.f, 0.f, 0.f, 0.f, 0.f, 0.f};
  v8f acc1 = {0.f, 0.f, 0.f, 0.f, 0.f, 0.f, 0.f, 0.f};
#pragma unroll
  for (int k0 = 0; k0 < K; k0 += 8) {
    v2f a0 = *(const v2f*)(Ap);
    v2f a1 = *(const v2f*)(Ap + 4);
    v2f b0, b1;
    b0.x = Bp0[0];
    b0.y = Bp1[0];
    b1.x = Bp0[(size_t)4 * N];
    b1.y = Bp1[(size_t)4 * N];
    acc0 = __builtin_amdgcn_wmma_f32_16x16x4_f32(
        false, a0, false, b0, (short)0, acc0, false, false);
    acc1 = __builtin_amdgcn_wmma_f32_16x16x4_f32(
        false, a1, false, b1, (short)0, acc1, false, false);
    Ap  += 8;
    Bp0 += bstep;
    Bp1 += bstep;
  }
  const v8f acc = acc0 + acc1;

  // C/D layout: reg r -> row (row0 + r + 8*half), col = col
  if (col < N) {
    const float bv = bias ? bias[col] : 0.f;
#pragma unroll
    for (int r = 0; r < 8; ++r) {
      const int rr = row0 + r + 8 * half;
      if (rr < M) {
        float v = acc[r] + bv;
        if (do_relu) v = fmaxf(v, 0.f);
        D[(size_t)rr * N + col] = v;
      }
    }
  }
}

// wa[j,h] = sum_k W[j, h*32+k] * att[h,k]   (collapsed dst-attention weights)
__global__ void collapse_att_kernel(const float* __restrict__ W,
                                    const float* __restrict__ att,
                                    float* __restrict__ wa) {
  const int t = threadIdx.x;           // 512 threads: j = t>>2, h = t&3
  const int j = t >> 2, h = t & 3;
  float s = 0.f;
#pragma unroll
  for (int k = 0; k < 32; ++k) s += W[j * 128 + h * 32 + k] * att[h * 32 + k];
  wa[j * 4 + h] = s;
}

// a_src[n,h] = sum_k HS[n, h*32+k] * att_src[h,k]
__global__ void att_src_kernel(const float* __restrict__ HS,
                               const float* __restrict__ att,
                               float* __restrict__ out, int N) {
  const int t = blockIdx.x * blockDim.x + threadIdx.x;
  if (t >= N * 4) return;
  const int n = t >> 2, h = t & 3;
  float s = 0.f;
#pragma unroll
  for (int k = 0; k < 32; ++k) s += HS[(size_t)n * 128 + h * 32 + k] * att[h * 32 + k];
  out[t] = s;
}

// a_dst[n,h] = sum_j X[n,j] * wa[j,h]
__global__ void att_dst_kernel(const float* __restrict__ X,
                               const float* __restrict__ wa,
                               float* __restrict__ out, int N) {
  const int t = blockIdx.x * blockDim.x + threadIdx.x;
  if (t >= N * 4) return;
  const int n = t >> 2, h = t & 3;
  float s = 0.f;
#pragma unroll 4
  for (int j = 0; j < 128; ++j) s += X[(size_t)n * 128 + j] * wa[j * 4 + h];
  out[t] = s;
}

__global__ void fill_u32_kernel(unsigned* __restrict__ p, unsigned v, int n) {
  const int t = blockIdx.x * blockDim.x + threadIdx.x;
  if (t < n) p[t] = v;
}

// Pass 1: segment max of leaky_relu(a_s[src]+a_d[dst]) via ordered-uint atomicMax
__global__ void edge_max_kernel(const int* __restrict__ src,
                                const int* __restrict__ dst,
                                const float* __restrict__ as,
                                const float* __restrict__ ad,
                                unsigned* __restrict__ m, int E) {
  const int e = blockIdx.x * blockDim.x + threadIdx.x;
  if (e >= E) return;
  const int s = src[e], d = dst[e];
#pragma unroll
  for (int h = 0; h < 4; ++h) {
    const float v = lrelu(as[s * 4 + h] + ad[d * 4 + h]);
    atomicMax(&m[d * 4 + h], enc_ord(v));
  }
}

// Pass 2: pe = exp(e - m[dst]); z[dst] += pe
__global__ void edge_sum_kernel(const int* __restrict__ src,
                                const int* __restrict__ dst,
                                const float* __restrict__ as,
                                const float* __restrict__ ad,
                                const unsigned* __restrict__ m,
                                float* __restrict__ z,
                                float* __restrict__ pe, int E) {
  const int e = blockIdx.x * blockDim.x + threadIdx.x;
  if (e >= E) return;
  const int s = src[e], d = dst[e];
#pragma unroll
  for (int h = 0; h < 4; ++h) {
    const float v = lrelu(as[s * 4 + h] + ad[d * 4 + h]);
    const float p = expf(v - dec_ord(m[d * 4 + h]));
    pe[e * 4 + h] = p;
    atomicAdd(&z[d * 4 + h], p);
  }
}

// Pass 3: acc[dst, j] += inv_cnt * (pe/z[dst]) * HS[src, j]   (j = 0..127)
__global__ void edge_scatter_kernel(const int* __restrict__ src,
                                    const int* __restrict__ dst,
                                    const float* __restrict__ pe,
                                    const float* __restrict__ z,
                                    const float* __restrict__ HS,
                                    float* __restrict__ acc,
                                    float inv_cnt, int E) {
  const long long t = (long long)blockIdx.x * blockDim.x + threadIdx.x;
  if (t >= (long long)E * 128) return;
  const int e = (int)(t >> 7);
  const int j = (int)(t & 127);
  const int h = j >> 5;
  const int s = src[e], d = dst[e];
  const float alpha = pe[e * 4 + h] / z[d * 4 + h];
  atomicAdd(&acc[(size_t)d * 128 + j], inv_cnt * alpha * HS[(size_t)s * 128 + j]);
}

// acc[n, j] += bias[j] * inv_cnt  (bias applies to every dst node)
__global__ void bias_add_kernel(float* __restrict__ acc,
                                const float* __restrict__ bias,
                                float inv_cnt, int N) {
  const int t = blockIdx.x * blockDim.x + threadIdx.x;
  if (t >= N * 128) return;
  acc[t] += bias[t & 127] * inv_cnt;
}

__global__ void relu_inplace_kernel(float* __restrict__ p, int n) {
  const int t = blockIdx.x * blockDim.x + threadIdx.x;
  if (t < n) p[t] = fmaxf(p[t], 0.f);
}

// ---------------------------------------------------------------------------
// Host orchestration
// ---------------------------------------------------------------------------
static const int NT_N[9] = {60000, 30000, 4000, 400, 15000, 30000, 8000, 20, 15000};
static const int NT_D[9] = {64, 64, 32, 16, 32, 32, 32, 16, 32};
// sorted edge keys: src node idx, dst node idx, edge count
static const int ET_S[19] = {0,0,0,0,0,0,0,0,0, 1, 2,2,2, 3, 4, 5, 6, 7, 8};
static const int ET_D[19] = {6,2,4,1,5,8,0,0,0, 0, 7,3,0, 2, 0, 0, 0, 2, 0};
static const int ET_E[19] = {50000,50000,50000,50000,50000,50000,150000,150000,150000,
                             50000, 4000,8000,50000, 8000, 50000,50000,50000, 4000, 50000};
// number of edge types targeting each dst node type (HeteroConv mean divisor)
static const int DST_CNT[9] = {9, 1, 3, 1, 1, 1, 1, 1, 1};

extern "C" void kernel_launch(void* const* d_in, const int* in_sizes, int n_in,
                              void* d_out, int out_size, void* d_ws, size_t ws_size,
                              hipStream_t stream) {
  (void)in_sizes; (void)n_in; (void)out_size; (void)ws_size;
  auto fin = [&](int i) { return (const float*)d_in[i]; };
  auto iin = [&](int i) { return (const int*)d_in[i]; };

  // Carve workspace (floats, 256B-aligned chunks)
  float* ws = (float*)d_ws;
  size_t off = 0;
  auto carve = [&](size_t nf) {
    float* p = ws + off;
    off += (nf + 63) & ~(size_t)63;
    return p;
  };
  float* xc[9];  // current node features [N,128]
  float* xn[9];  // next-layer accumulators [N,128]
  for (int nt = 0; nt < 9; ++nt) xc[nt] = carve((size_t)NT_N[nt] * 128);
  for (int nt = 0; nt < 9; ++nt) xn[nt] = carve((size_t)NT_N[nt] * 128);
  float*    HS  = carve((size_t)60000 * 128);  // per-edge-type src projection; reused as head h1
  float*    a_s = carve((size_t)60000 * 4);
  float*    a_d = carve((size_t)60000 * 4);
  float*    wa  = carve(128 * 4);
  unsigned* mb  = (unsigned*)carve((size_t)60000 * 4);
  float*    zb  = carve((size_t)60000 * 4);
  float*    pe  = carve((size_t)150000 * 4);

  auto gemm = [&](const float* A, const float* B, const float* bias, float* D,
                  int M, int N, int K, int relu) {
    dim3 g((unsigned)((M + 15) / 16), (unsigned)((N + 15) / 16));
    switch (K) {
      case 16:
        gemm_wmma_f32_kernel<16><<<g, dim3(32), 0, stream>>>(A, B, bias, D, M, N, relu);
        break;
      case 32:
        gemm_wmma_f32_kernel<32><<<g, dim3(32), 0, stream>>>(A, B, bias, D, M, N, relu);
        break;
      case 64:
        gemm_wmma_f32_kernel<64><<<g, dim3(32), 0, stream>>>(A, B, bias, D, M, N, relu);
        break;
      default:
        gemm_wmma_f32_kernel<128><<<g, dim3(32), 0, stream>>>(A, B, bias, D, M, N, relu);
        break;
    }
  };
  auto fill = [&](void* p, unsigned v, int n) {
    fill_u32_kernel<<<(n + 255) / 256, 256, 0, stream>>>((unsigned*)p, v, n);
  };

  // --- Encoders: x[nt] = feats[nt] @ encW + encB ---
  for (int nt = 0; nt < 9; ++nt)
    gemm(fin(38 + nt), fin(199 + 2 * nt), fin(200 + 2 * nt), xc[nt],
         NT_N[nt], 128, NT_D[nt], 0);

  // --- 2 hetero GAT layers ---
  for (int li = 0; li < 2; ++li) {
    for (int nt = 0; nt < 9; ++nt) fill(xn[nt], 0u, NT_N[nt] * 128);
    for (int et = 0; et < 19; ++et) {
      const int s = ET_S[et], d = ET_D[et], E = ET_E[et];
      const int Ns = NT_N[s], Nd = NT_N[d];
      const int pb = 47 + li * 76 + et * 4;
      const float* W    = fin(pb);
      const float* attD = fin(pb + 1);
      const float* attS = fin(pb + 2);
      const float* bias = fin(pb + 3);
      const int* esrc = iin(2 * et);
      const int* edst = iin(2 * et + 1);
      const float inv_cnt = 1.0f / (float)DST_CNT[d];

      gemm(xc[s], W, nullptr, HS, Ns, 128, 128, 0);                 // HS = x_src @ W
      collapse_att_kernel<<<1, 512, 0, stream>>>(W, attD, wa);      // wa = W . att_dst
      att_src_kernel<<<(Ns * 4 + 255) / 256, 256, 0, stream>>>(HS, attS, a_s, Ns);
      att_dst_kernel<<<(Nd * 4 + 255) / 256, 256, 0, stream>>>(xc[d], wa, a_d, Nd);
      fill(mb, 0x007FFFFFu, Nd * 4);   // enc_ord(-inf)
      fill(zb, 0u, Nd * 4);
      edge_max_kernel<<<(E + 255) / 256, 256, 0, stream>>>(esrc, edst, a_s, a_d, mb, E);
      edge_sum_kernel<<<(E + 255) / 256, 256, 0, stream>>>(esrc, edst, a_s, a_d, mb, zb, pe, E);
      const long long tot = (long long)E * 128;
      edge_scatter_kernel<<<(unsigned)((tot + 255) / 256), 256, 0, stream>>>(
          esrc, edst, pe, zb, HS, xn[d], inv_cnt, E);
      bias_add_kernel<<<(Nd * 128 + 255) / 256, 256, 0, stream>>>(xn[d], bias, inv_cnt, Nd);
    }
    if (li == 0)
      for (int nt = 0; nt < 9; ++nt) {
        const int n = NT_N[nt] * 128;
        relu_inplace_kernel<<<(n + 255) / 256, 256, 0, stream>>>(xn[nt], n);
      }
    for (int nt = 0; nt < 9; ++nt) { float* t = xc[nt]; xc[nt] = xn[nt]; xn[nt] = t; }
  }

  // --- Head MLP on alert nodes ---
  float* h1 = HS;  // HS no longer needed; 60000*64 fits in its slab
  gemm(xc[0], fin(217), fin(218), h1, 60000, 64, 128, 1);
  gemm(h1, fin(219), fin(220), (float*)d_out, 60000, 10, 64, 0);
}